// Original_Conv_layer_60842506715663
// MI455X (gfx1250) — compile-verified
//
#include <hip/hip_runtime.h>
#include <hip/hip_bf16.h>

typedef __attribute__((ext_vector_type(2))) float v2f;
typedef __attribute__((ext_vector_type(8))) float v8f;

#define BS     4
#define NV     2048
#define NN     20
#define IN_CH  128
#define OUT_CH 128
#define NS     4
#define SXO    (NS * OUT_CH)          // 512
#define NCOL   ((NS + 1) * OUT_CH)    // 640
#define NROW   (BS * NV)              // 8192
#define EPSF   1e-12f

// ---------------------------------------------------------------------------
// Kernel 1: feature_out = feature_map @ weights + bias  via V_WMMA_F32_16X16X4_F32
// M=8192, K=128, N=640. One 16x16 tile per wave; 8 waves/block; grid exact.
// ---------------------------------------------------------------------------
__global__ __launch_bounds__(256)
void gemm_wmma_f32_kernel(const float* __restrict__ fm,
                          const float* __restrict__ w,
                          const float* __restrict__ bias,
                          float* __restrict__ fout) {
    const int lane  = threadIdx.x & 31;
    const int wave  = threadIdx.x >> 5;
    const int gwave = blockIdx.x * 8 + wave;

    const int ntiles_n = NCOL / 16;          // 40
    const int tm = gwave / ntiles_n;         // 0..511
    const int tn = gwave % ntiles_n;         // 0..39

    const int half = lane >> 4;              // 0: lanes 0-15, 1: lanes 16-31
    const int l15  = lane & 15;
    const int r0   = tm * 16;
    const int c0   = tn * 16;

    v8f acc = {};
    // A (16x4 f32): lane M = l15; VGPR0/1 = K = kk + 2*half + {0,1}
    // B (4x16 f32): lane N = l15; VGPR0/1 = K = kk + 2*half + {0,1} rows of W
    for (int kk = 0; kk < IN_CH; kk += 4) {
        const float* arow = fm + (size_t)(r0 + l15) * IN_CH + (kk + 2 * half);
        v2f a;
        a.x = arow[0];
        a.y = arow[1];

        const float* brow = w + (size_t)(kk + 2 * half) * NCOL + c0 + l15;
        v2f b;
        b.x = brow[0];
        b.y = brow[NCOL];

        acc = __builtin_amdgcn_wmma_f32_16x16x4_f32(
            /*neg_a=*/false, a, /*neg_b=*/false, b,
            /*c_mod=*/(short)0, acc, /*reuse_a=*/false, /*reuse_b=*/false);
    }

    // C/D 16x16 f32 layout: VGPR r, lanes 0-15 -> M=r, lanes 16-31 -> M=r+8; N=l15
    const float bv = bias[c0 + l15];
#pragma unroll
    for (int r = 0; r < 8; ++r) {
        const int row = r0 + r + half * 8;
        fout[(size_t)row * NCOL + c0 + l15] = acc[r] + bv;
    }
}

// ---------------------------------------------------------------------------
// Kernel 2: fused gather / theta / max-over-neighbors / s-group reduce.
// One 256-thread block per (b,v). Gathers of feature_out rows hit L2 (21MB).
// ---------------------------------------------------------------------------
__global__ __launch_bounds__(256)
void fused_gconv_kernel(const int*   __restrict__ nidx,
                        const float* __restrict__ verts,
                        const float* __restrict__ fout,
                        const float* __restrict__ dirs,      // (3, 512)
                        const float* __restrict__ dw,        // (1, 512)
                        float*       __restrict__ out) {
    const int bv = blockIdx.x;               // b*NV + v
    const int b  = bv / NV;

    __shared__ float s_dn[NN][3];
    __shared__ float s_dist[NN];
    __shared__ int   s_idx[NN];
    __shared__ float s_maxd;
    __shared__ float s_vals[SXO];

    const int tid = threadIdx.x;

    // Phase A: neighbor geometry (threads 0..19)
    if (tid < NN) {
        const int j   = tid;
        const int idx = nidx[(size_t)bv * NN + j];
        s_idx[j] = idx;

        const float cx = verts[(size_t)bv * 3 + 0];
        const float cy = verts[(size_t)bv * 3 + 1];
        const float cz = verts[(size_t)bv * 3 + 2];

        const size_t np = ((size_t)b * NV + idx) * 3;
        const float dx = verts[np + 0] - cx;
        const float dy = verts[np + 1] - cy;
        const float dz = verts[np + 2] - cz;

        const float dist = sqrtf(dx * dx + dy * dy + dz * dz);
        const float inv  = 1.0f / fmaxf(dist, EPSF);
        s_dn[j][0] = dx * inv;
        s_dn[j][1] = dy * inv;
        s_dn[j][2] = dz * inv;
        s_dist[j]  = dist;
    }
    __syncthreads();
    if (tid == 0) {
        float md = s_dist[0];
#pragma unroll
        for (int j = 1; j < NN; ++j) md = fmaxf(md, s_dist[j]);
        s_maxd = md;
    }
    __syncthreads();
    const float max_dist = s_maxd;

    // Phase B: each thread owns channels m = tid and tid+256 of the 512.
#pragma unroll
    for (int mm = 0; mm < 2; ++mm) {
        const int m = tid + mm * 256;

        // sdn column m: normalize directions[:, m]
        const float sx  = dirs[0 * SXO + m];
        const float sy  = dirs[1 * SXO + m];
        const float sz  = dirs[2 * SXO + m];
        const float nrm = sqrtf(sx * sx + sy * sy + sz * sz);
        const float inv = 1.0f / fmaxf(nrm, EPSF);
        const float ux = sx * inv, uy = sy * inv, uz = sz * inv;

        float acc = -INFINITY;
#pragma unroll 4
        for (int j = 0; j < NN; ++j) {
            const float dot   = s_dn[j][0] * ux + s_dn[j][1] * uy + s_dn[j][2] * uz;
            const float theta = fmaxf(dot, 0.0f);
            const size_t frow = ((size_t)b * NV + s_idx[j]) * NCOL + OUT_CH + m;
            const float fsup  = fout[frow];
            acc = fmaxf(acc, theta * fsup);
        }
        const float dist_feat = fmaxf(max_dist * dw[m], 0.0f);
        s_vals[m] = acc + dist_feat;
    }
    __syncthreads();

    // Final: sum the 4 s-groups + feature_center
    if (tid < OUT_CH) {
        const float fc = fout[(size_t)bv * NCOL + tid];
        float sum = s_vals[tid] + s_vals[OUT_CH + tid] +
                    s_vals[2 * OUT_CH + tid] + s_vals[3 * OUT_CH + tid];
        out[(size_t)bv * OUT_CH + tid] = fc + sum;
    }
}

extern "C" void kernel_launch(void* const* d_in, const int* in_sizes, int n_in,
                              void* d_out, int out_size, void* d_ws, size_t ws_size,
                              hipStream_t stream) {
    const int*   nidx  = (const int*)  d_in[0];   // (4,2048,20)
    const float* verts = (const float*)d_in[1];   // (4,2048,3)
    const float* fm    = (const float*)d_in[2];   // (4,2048,128)
    const float* w     = (const float*)d_in[3];   // (128,640)
    const float* bias  = (const float*)d_in[4];   // (640,)
    const float* dirs  = (const float*)d_in[5];   // (3,512)
    const float* dw    = (const float*)d_in[6];   // (1,512)
    float*       out   = (float*)d_out;           // (4,2048,128)
    float*       fout  = (float*)d_ws;            // (8192,640) scratch

    // GEMM: 512*40 = 20480 tiles, 1 tile/wave, 8 waves/block -> 2560 blocks
    const int gemm_blocks = (NROW / 16) * (NCOL / 16) / 8;
    gemm_wmma_f32_kernel<<<gemm_blocks, 256, 0, stream>>>(fm, w, bias, fout);

    // Fused stage: one block per (b,v)
    fused_gconv_kernel<<<NROW, 256, 0, stream>>>(nidx, verts, fout, dirs, dw, out);
}